// SimpleGCN_28965259444794
// MI455X (gfx1250) — compile-verified
//
#include <hip/hip_runtime.h>
#include <hip/hip_bf16.h>

// ---------------------------------------------------------------------------
// SimpleGCN on MI455X (gfx1250, wave32).
//  - f32 WMMA (V_WMMA_F32_16X16X4_F32) for the dense GEMMs (exact precision).
//  - L2-resident gather + global_atomic_add_f32 scatter for aggregation.
// ---------------------------------------------------------------------------

typedef __attribute__((ext_vector_type(2))) float v2f;
typedef __attribute__((ext_vector_type(8))) float v8f;

// Only check on the device pass: __has_builtin() for amdgcn target builtins
// returns 0 during hipcc's host (x86) compilation.
#if defined(__HIP_DEVICE_COMPILE__)
#if !__has_builtin(__builtin_amdgcn_wmma_f32_16x16x4_f32)
#error "missing __builtin_amdgcn_wmma_f32_16x16x4_f32 on this toolchain"
#endif
#endif

#define NODES 100000
#define EDGES 1600000
#define INDIM 256
#define HID   128

// --------------------------- degree / normalization ------------------------

__global__ __launch_bounds__(256) void deg_init(float* __restrict__ deg) {
  int i = blockIdx.x * 256 + threadIdx.x;
  if (i < NODES) deg[i] = 1.0f;            // self-loop contributes 1
}

__global__ __launch_bounds__(256) void deg_count(const int* __restrict__ dst,
                                                 float* __restrict__ deg) {
  int e = blockIdx.x * 256 + threadIdx.x;
  if (e < EDGES) atomicAdd(&deg[dst[e]], 1.0f);
}

__global__ __launch_bounds__(256) void deg_finalize(float* __restrict__ dinv) {
  int i = blockIdx.x * 256 + threadIdx.x;
  if (i < NODES) {
    float d = dinv[i];                     // >= 1 because of self-loop
    dinv[i] = (d > 0.0f) ? rsqrtf(d) : 0.0f;
  }
}

// --------------------------- WMMA f32 GEMM ---------------------------------
// D[M x HID] = act(A[M x K] * W[K x HID] (+ bias)).  One block = 16 rows,
// 8 waves, each wave owns a 16x16 output tile (HID = 8*16 = 128 cols).

template <int K, bool BIAS_RELU>
__global__ __launch_bounds__(256) void gemm_wmma_f32(
    const float* __restrict__ A, const float* __restrict__ W,
    const float* __restrict__ bias, float* __restrict__ D) {
  constexpr int LDA = K + 1;               // LDS pad -> conflict-free strided reads
  __shared__ float Als[16 * LDA];

  const int tid = threadIdx.x;
  const int m0  = blockIdx.x * 16;

  // Stage the 16 x K strip of A into LDS (coalesced along k).
  for (int idx = tid; idx < 16 * K; idx += 256) {
    int r = idx / K;
    int k = idx - r * K;
    Als[r * LDA + k] = A[(size_t)(m0 + r) * K + k];
  }
  __syncthreads();

  const int wave = tid >> 5;
  const int lane = tid & 31;
  const int hi   = lane >> 4;              // 0: K even pair, 1: K odd pair half
  const int l16  = lane & 15;
  const int n0   = wave * 16;              // this wave's 16 output columns

  v8f c = {};
  const float* arow = Als + l16 * LDA;     // A row for this lane (M = l16)

  #pragma unroll 4
  for (int kk = 0; kk < K; kk += 4) {
    // A fragment, 16x4 f32: VGPR0 = K=kk(+2), VGPR1 = K=kk+1(+3), split at lane 16.
    v2f a, b;
    const int ak = kk + hi * 2;
    a.x = arow[ak];
    a.y = arow[ak + 1];
    // B fragment, 4x16 f32: lane = column, VGPR0/1 = K pair, split at lane 16.
    const float* wcol = W + (size_t)ak * HID + n0 + l16;
    b.x = wcol[0];
    b.y = wcol[HID];
    c = __builtin_amdgcn_wmma_f32_16x16x4_f32(false, a, false, b,
                                              (short)0, c, false, false);
  }

  // D layout: VGPR v -> row (v + hi*8), column = n0 + l16.
  const int col = n0 + l16;
  float bval = BIAS_RELU ? bias[col] : 0.0f;
  #pragma unroll
  for (int v = 0; v < 8; ++v) {
    float val = c[v];
    if (BIAS_RELU) val = fmaxf(val + bval, 0.0f);
    D[(size_t)(m0 + hi * 8 + v) * HID + col] = val;
  }
}

// --------------------------- aggregation -----------------------------------
// out[i,:] = b + dinv[i]^2 * hw[i,:]   (bias + self-loop message, no atomics)

__global__ __launch_bounds__(256) void agg_init(
    const float* __restrict__ hw, const float* __restrict__ dinv,
    const float* __restrict__ bias, float* __restrict__ out) {
  int t = blockIdx.x * 256 + threadIdx.x;  // one float4 per thread
  if (t >= NODES * 32) return;
  int i  = t >> 5;
  int c4 = (t & 31) * 4;
  float w = dinv[i];
  w *= w;
  const float4 m = *(const float4*)(hw + (size_t)i * HID + c4);
  float4 o;
  o.x = bias[c4 + 0] + w * m.x;
  o.y = bias[c4 + 1] + w * m.y;
  o.z = bias[c4 + 2] + w * m.z;
  o.w = bias[c4 + 3] + w * m.w;
  *(float4*)(out + (size_t)i * HID + c4) = o;
}

// One wave per edge; lane owns a float4 slice of the 128-wide message.
// Gather hits L2 (51 MB feature matrix resident); scatter = f32 L2 atomics.
__global__ __launch_bounds__(256) void gcn_scatter(
    const int* __restrict__ src, const int* __restrict__ dst,
    const float* __restrict__ dinv, const float* __restrict__ hw,
    float* __restrict__ out) {
  int g    = blockIdx.x * 256 + threadIdx.x;
  int e    = g >> 5;
  int lane = g & 31;
  if (e >= EDGES) return;
  int s = src[e];
  int d = dst[e];
  float w = dinv[s] * dinv[d];
  const float4 m = *(const float4*)(hw + (size_t)s * HID + lane * 4);
  float* op = out + (size_t)d * HID + lane * 4;
  atomicAdd(op + 0, w * m.x);
  atomicAdd(op + 1, w * m.y);
  atomicAdd(op + 2, w * m.z);
  atomicAdd(op + 3, w * m.w);
}

// --------------------------- head (128 -> 2) -------------------------------

__global__ __launch_bounds__(256) void head_kernel(
    const float* __restrict__ h, const float* __restrict__ Wh,
    const float* __restrict__ bh, float* __restrict__ out) {
  int i = blockIdx.x * 256 + threadIdx.x;
  if (i >= NODES) return;
  float a0 = bh[0], a1 = bh[1];
  const float* hr = h + (size_t)i * HID;
  #pragma unroll 8
  for (int k = 0; k < HID; ++k) {
    float hv = hr[k];
    a0 = fmaf(hv, Wh[k * 2 + 0], a0);
    a1 = fmaf(hv, Wh[k * 2 + 1], a1);
  }
  out[(size_t)i * 2 + 0] = a0;
  out[(size_t)i * 2 + 1] = a1;
}

// --------------------------- launcher --------------------------------------

extern "C" void kernel_launch(void* const* d_in, const int* in_sizes, int n_in,
                              void* d_out, int out_size, void* d_ws, size_t ws_size,
                              hipStream_t stream) {
  (void)in_sizes; (void)n_in; (void)out_size; (void)ws_size;

  const int*   edge = (const int*)d_in[0];      // [2, E] flat
  const float* x    = (const float*)d_in[1];
  const float* Wi   = (const float*)d_in[2];
  const float* bi   = (const float*)d_in[3];
  const float* W1   = (const float*)d_in[4];
  const float* b1   = (const float*)d_in[5];
  const float* W2   = (const float*)d_in[6];
  const float* b2   = (const float*)d_in[7];
  const float* Wh   = (const float*)d_in[8];
  const float* bh   = (const float*)d_in[9];
  float*       out  = (float*)d_out;

  const int* src = edge;                        // edge_index[0]
  const int* dst = edge + EDGES;                // edge_index[1]

  // Workspace: dinv[N] | hA[N*HID] | hW[N*HID]  (~103 MB)
  float* dinv = (float*)d_ws;
  float* hA   = dinv + NODES;
  float* hW   = hA + (size_t)NODES * HID;

  const int nb_node = (NODES + 255) / 256;
  const int nb_edge = (EDGES + 255) / 256;
  const int nb_gemm = NODES / 16;               // 100000 % 16 == 0
  const int nb_vec  = (NODES * 32) / 256;       // agg_init, exact
  const int nb_scat = (EDGES * 32) / 256;       // one wave per edge, exact

  // Normalization: deg (with self-loop) -> dinv = deg^-1/2
  deg_init    <<<nb_node, 256, 0, stream>>>(dinv);
  deg_count   <<<nb_edge, 256, 0, stream>>>(dst, dinv);
  deg_finalize<<<nb_node, 256, 0, stream>>>(dinv);

  // NodeInitializer: hA = relu(x @ Wi + bi)
  gemm_wmma_f32<INDIM, true><<<nb_gemm, 256, 0, stream>>>(x, Wi, bi, hA);

  // GCN layer 1: hW = hA @ W1 ; hA = scatter(norm * hW) + b1
  gemm_wmma_f32<HID, false><<<nb_gemm, 256, 0, stream>>>(hA, W1, nullptr, hW);
  agg_init   <<<nb_vec,  256, 0, stream>>>(hW, dinv, b1, hA);
  gcn_scatter<<<nb_scat, 256, 0, stream>>>(src, dst, dinv, hW, hA);

  // GCN layer 2
  gemm_wmma_f32<HID, false><<<nb_gemm, 256, 0, stream>>>(hA, W2, nullptr, hW);
  agg_init   <<<nb_vec,  256, 0, stream>>>(hW, dinv, b2, hA);
  gcn_scatter<<<nb_scat, 256, 0, stream>>>(src, dst, dinv, hW, hA);

  // Head: out = hA @ Wh + bh
  head_kernel<<<nb_node, 256, 0, stream>>>(hA, Wh, bh, out);
}